// GATConv_13091060319062
// MI455X (gfx1250) — compile-verified
//
#include <hip/hip_runtime.h>
#include <hip/hip_bf16.h>

// ---------------------------------------------------------------------------
// GATConv forward for gfx1250 (MI455X, wave32).
// N=100000 nodes, E=1600000 edges, D_IN=128, H=4, D=32 (H*D=128).
//
// Pipeline:
//   1) gemm_ft_wmma   : ft = feat @ W^T        (v_wmma_f32_16x16x4_f32)
//   2) attn_logits    : el/er per node          (wave shuffle reduction)
//   3) init_node      : rst=0, s=0, m=-inf
//   4) edge_max       : segment max of LeakyReLU logits (encoded u32 atomicMax)
//   5) edge_aggregate : rst[dst] += exp(e-m)*ft[src];  s[dst] += exp(e-m)
//   6) normalize      : rst = rst / max(s,1e-38) + bias
// ---------------------------------------------------------------------------

typedef __attribute__((ext_vector_type(2))) float v2f;
typedef __attribute__((ext_vector_type(8))) float v8f;

#define NEG_SLOPE 0.2f
#define HD 128          // H*D == D_IN
#define NHEAD 4

// ----- order-preserving float <-> uint encoding for atomic max --------------
__device__ __forceinline__ unsigned int encodeF32(float f) {
    unsigned int b = __float_as_uint(f);
    return (b & 0x80000000u) ? ~b : (b | 0x80000000u);
}
__device__ __forceinline__ float decodeF32(unsigned int u) {
    unsigned int b = (u & 0x80000000u) ? (u & 0x7FFFFFFFu) : ~u;
    return __uint_as_float(b);
}
#define ENC_NEG_INF 0x007FFFFFu   // encodeF32(-inf)

// ---------------------------------------------------------------------------
// Kernel 1: ft = feat @ W^T  via V_WMMA_F32_16X16X4_F32.
// One block = 16 rows of ft; wave w computes the 16-col tile starting at w*16.
// W (128x128, 64KB) staged transposed into LDS: ldsWt[k*128+n] = W[n*128+k].
// ---------------------------------------------------------------------------
__global__ __launch_bounds__(256) void gemm_ft_wmma(
        const float* __restrict__ feat, const float* __restrict__ W,
        float* __restrict__ ft, int Nn) {
    __shared__ float ldsWt[HD * HD];      // exactly 64 KB

    const int t = threadIdx.x;
    // coalesced global read of W, transposed write to LDS
    for (int idx = t; idx < HD * HD; idx += 256) {
        int n = idx >> 7;        // row of W  (output feature)
        int k = idx & 127;       // col of W  (input feature)
        ldsWt[k * HD + n] = W[idx];
    }
    __syncthreads();

    const int lane = t & 31;
    const int wave = t >> 5;           // 8 waves -> 8 column tiles (8*16 = 128)
    const int m    = lane & 15;        // row within 16-tile / col for B
    const int kk   = lane >> 4;        // K-half select (K=0,1 vs K=2,3)
    const int col_base = wave * 16;
    const int row_base = blockIdx.x * 16;

    int arow = row_base + m;
    if (arow >= Nn) arow = Nn - 1;     // clamp loads; stores guarded below
    const float* __restrict__ aptr = feat + (size_t)arow * HD;

    v8f acc = {};
    #pragma unroll 4
    for (int k = 0; k < HD; k += 4) {
        // A tile 16x4 f32: lane m holds (K = 2*kk, 2*kk+1) in regs 0/1
        v2f a = *(const v2f*)(aptr + k + 2 * kk);
        // B tile 4x16 f32: B[kl][n] = W^T[kl][n] = ldsWt[kl*128 + n]
        v2f b;
        b.x = ldsWt[(k + 2 * kk)     * HD + col_base + m];
        b.y = ldsWt[(k + 2 * kk + 1) * HD + col_base + m];
        acc = __builtin_amdgcn_wmma_f32_16x16x4_f32(
                /*neg_a=*/false, a, /*neg_b=*/false, b,
                /*c_mod=*/(short)0, acc, /*reuse_a=*/false, /*reuse_b=*/false);
    }

    // C/D layout: reg r -> row r + 8*kk, col = lane&15
    float* __restrict__ outp =
        ft + (size_t)(row_base + 8 * kk) * HD + col_base + m;
    if (row_base + 16 <= Nn) {
        // full tile: 8 plain stores at fixed 512B strides off one base
        #pragma unroll
        for (int r = 0; r < 8; ++r)
            outp[(size_t)r * HD] = acc[r];
    } else {
        #pragma unroll
        for (int r = 0; r < 8; ++r) {
            int row = row_base + r + 8 * kk;
            if (row < Nn) outp[(size_t)r * HD] = acc[r];
        }
    }
}

// ---------------------------------------------------------------------------
// Kernel 2: per-node attention logits. One wave per node, lane = channel d.
// ---------------------------------------------------------------------------
__global__ __launch_bounds__(256) void attn_logits(
        const float* __restrict__ ft,
        const float* __restrict__ attn_l, const float* __restrict__ attn_r,
        float* __restrict__ el, float* __restrict__ er, int Nn) {
    const int node = (int)((blockIdx.x * (unsigned)blockDim.x + threadIdx.x) >> 5);
    const int lane = threadIdx.x & 31;
    if (node >= Nn) return;
    const float* __restrict__ f = ft + (size_t)node * HD;
    #pragma unroll
    for (int h = 0; h < NHEAD; ++h) {
        float v = f[h * 32 + lane];
        float l = v * attn_l[h * 32 + lane];
        float r = v * attn_r[h * 32 + lane];
        #pragma unroll
        for (int off = 16; off > 0; off >>= 1) {
            l += __shfl_xor(l, off, 32);
            r += __shfl_xor(r, off, 32);
        }
        if (lane == 0) {
            el[node * NHEAD + h] = l;
            er[node * NHEAD + h] = r;
        }
    }
}

// ---------------------------------------------------------------------------
// Kernel 3: zero rst (unnormalized accumulator); zero s; segment-max = -inf.
// ---------------------------------------------------------------------------
__global__ __launch_bounds__(256) void init_node(
        float* __restrict__ rst,
        float* __restrict__ s, unsigned int* __restrict__ menc, int Nn) {
    const long long i = blockIdx.x * 256LL + threadIdx.x;
    if (i < (long long)Nn * HD) rst[i] = 0.0f;
    if (i < (long long)Nn * NHEAD) { s[i] = 0.0f; menc[i] = ENC_NEG_INF; }
}

// ---------------------------------------------------------------------------
// Kernel 4: edge logits + segment max (encoded u32 atomicMax).
// ---------------------------------------------------------------------------
__global__ __launch_bounds__(256) void edge_max(
        const int* __restrict__ src, const int* __restrict__ dst,
        const float* __restrict__ el, const float* __restrict__ er,
        unsigned int* __restrict__ menc, int Ee) {
    const int e = blockIdx.x * 256 + threadIdx.x;
    if (e >= Ee) return;
    const int sN = src[e], dN = dst[e];
    const float4 l = *(const float4*)(el + (size_t)sN * NHEAD);
    const float4 r = *(const float4*)(er + (size_t)dN * NHEAD);
    const float* lp = &l.x; const float* rp = &r.x;
    #pragma unroll
    for (int h = 0; h < NHEAD; ++h) {
        float v = lp[h] + rp[h];
        v = v > 0.0f ? v : NEG_SLOPE * v;
        atomicMax(&menc[(size_t)dN * NHEAD + h], encodeF32(v));
    }
}

// ---------------------------------------------------------------------------
// Kernel 5 (fused sum+aggregate): one wave per edge.
//   ex = exp(LeakyReLU(el[src]+er[dst]) - m[dst])
//   rst[dst] += ex * ft[src]   (unnormalized; divide happens in normalize)
//   s[dst]   += ex             (one lane per 8-lane head group)
// Lane owns 4 contiguous channels of head h = lane>>3.
// ---------------------------------------------------------------------------
__global__ __launch_bounds__(256) void edge_aggregate(
        const int* __restrict__ src, const int* __restrict__ dst,
        const float* __restrict__ el, const float* __restrict__ er,
        const unsigned int* __restrict__ menc, float* __restrict__ s,
        const float* __restrict__ ft, float* __restrict__ rst, int Ee) {
    const long long gtid = blockIdx.x * 256LL + threadIdx.x;
    const int e    = (int)(gtid >> 5);
    const int lane = threadIdx.x & 31;
    if (e >= Ee) return;
    const int sN = src[e], dN = dst[e];
    const int h  = lane >> 3;                       // 8 lanes per head

    float v = el[(size_t)sN * NHEAD + h] + er[(size_t)dN * NHEAD + h];
    v = v > 0.0f ? v : NEG_SLOPE * v;
    const float m  = decodeF32(menc[(size_t)dN * NHEAD + h]);
    const float ex = __expf(v - m);

    if ((lane & 7) == 0)
        unsafeAtomicAdd(&s[(size_t)dN * NHEAD + h], ex);

    const float4 fv = *(const float4*)(ft + (size_t)sN * HD + lane * 4);
    float* out = rst + (size_t)dN * HD + lane * 4;
    unsafeAtomicAdd(out + 0, fv.x * ex);
    unsafeAtomicAdd(out + 1, fv.y * ex);
    unsafeAtomicAdd(out + 2, fv.z * ex);
    unsafeAtomicAdd(out + 3, fv.w * ex);
}

// ---------------------------------------------------------------------------
// Kernel 6: rst = rst / max(s,1e-38) + bias   (coalesced, in place)
// ---------------------------------------------------------------------------
__global__ __launch_bounds__(256) void normalize(
        float* __restrict__ rst, const float* __restrict__ s,
        const float* __restrict__ bias, int Nn) {
    const long long i = blockIdx.x * 256LL + threadIdx.x;
    if (i >= (long long)Nn * HD) return;
    const int j = (int)(i & (HD - 1));   // h*32 + d
    const int n = (int)(i >> 7);
    float den = s[(size_t)n * NHEAD + (j >> 5)];
    den = den > 1e-38f ? den : 1e-38f;
    rst[i] = rst[i] / den + bias[j];
}

// ---------------------------------------------------------------------------
extern "C" void kernel_launch(void* const* d_in, const int* in_sizes, int n_in,
                              void* d_out, int out_size, void* d_ws, size_t ws_size,
                              hipStream_t stream) {
    const float* feat   = (const float*)d_in[0];
    const int*   src    = (const int*)  d_in[1];
    const int*   dst    = (const int*)  d_in[2];
    const float* W      = (const float*)d_in[3];
    const float* attn_l = (const float*)d_in[4];
    const float* attn_r = (const float*)d_in[5];
    const float* bias   = (const float*)d_in[6];
    float*       rst    = (float*)d_out;

    const int Nn = in_sizes[0] / HD;   // 100000
    const int Ee = in_sizes[1];        // 1600000

    // workspace carve (all chunks 16B aligned)
    char* ws = (char*)d_ws;
    float* ft = (float*)ws;               ws += (size_t)Nn * HD * sizeof(float);
    float* el = (float*)ws;               ws += (size_t)Nn * NHEAD * sizeof(float);
    float* er = (float*)ws;               ws += (size_t)Nn * NHEAD * sizeof(float);
    float* s  = (float*)ws;               ws += (size_t)Nn * NHEAD * sizeof(float);
    unsigned int* menc = (unsigned int*)ws;

    // 1) projection GEMM (WMMA f32)
    gemm_ft_wmma<<<(Nn + 15) / 16, 256, 0, stream>>>(feat, W, ft, Nn);
    // 2) per-node logits
    attn_logits<<<(Nn * 32 + 255) / 256, 256, 0, stream>>>(ft, attn_l, attn_r, el, er, Nn);
    // 3) init accumulators
    init_node<<<((long long)Nn * HD + 255) / 256, 256, 0, stream>>>(rst, s, menc, Nn);
    // 4) segment max
    edge_max<<<(Ee + 255) / 256, 256, 0, stream>>>(src, dst, el, er, menc, Ee);
    // 5) fused exp-weighted aggregation + denominator (wave per edge)
    edge_aggregate<<<(Ee * 32LL + 255) / 256, 256, 0, stream>>>(
            src, dst, el, er, menc, s, ft, rst, Ee);
    // 6) normalize + bias
    normalize<<<((long long)Nn * HD + 255) / 256, 256, 0, stream>>>(rst, s, bias, Nn);
}